// RCNN_87917980549798
// MI455X (gfx1250) — compile-verified
//
#include <hip/hip_runtime.h>

// RCNN loss (classification CE + masked smooth-L1) for MI455X (gfx1250, wave32).
//
// Memory-bound streaming reduction. One-hot structure lets us read only the
// 8 delta floats at the label class per row (~61 MB total traffic -> ~3us at
// 23.3 TB/s). Wave-cooperative layout: one row per wave iteration, lanes 0-20
// load the 21 scores coalesced; butterfly shuffles compute the row sum and
// locate the label lane; lanes 0-7 fetch output/target deltas in two aligned
// 16B bursts. Final per-wave reduction of the two accumulators is done with
// V_WMMA_F32_16X16X4_F32 against an all-ones B matrix (layout-invariant,
// exact f32 accumulate): D[m][n] = v_m + v_{m+16}, so sum of the 8 D VGPRs
// plus one shfl_xor(16) is the exact 32-lane sum.

#define C_CLS 21
#define D_DIM 84
#define EPS_F 1e-7f

typedef __attribute__((ext_vector_type(2))) float v2f;
typedef __attribute__((ext_vector_type(8))) float v8f;

// Exact 32-lane f32 sum via WMMA (B = ones => B-matrix VGPR layout irrelevant).
// Requires EXEC all-ones (call only from fully converged code).
__device__ __forceinline__ float wave_sum_wmma(float v) {
    v2f a; a[0] = v;    a[1] = 0.0f;   // A[m][0]=v(lane m), A[m][2]=v(lane m+16), K=1,3 are zero
    v2f b; b[0] = 1.0f; b[1] = 1.0f;   // all-ones 4x16
    v8f c = {};
    v8f d = __builtin_amdgcn_wmma_f32_16x16x4_f32(false, a, false, b, (short)0, c, false, false);
    // lane n<16: d[j] = D[j][n]   = v_j   + v_{j+16}
    // lane n>=16: d[j] = D[j+8][n] = v_{j+8} + v_{j+24}
    float s = ((d[0] + d[1]) + (d[2] + d[3])) + ((d[4] + d[5]) + (d[6] + d[7]));
    s += __shfl_xor(s, 16);            // all lanes now hold the full 32-lane sum
    return s;
}

__global__ void rcnn_loss_partial(const float* __restrict__ tgt_d,
                                  const float* __restrict__ tgt_s,
                                  const float* __restrict__ out_d,
                                  const float* __restrict__ out_s,
                                  float* __restrict__ partial,
                                  int nRows) {
    const int lane          = threadIdx.x & 31;
    const int waveInBlock   = threadIdx.x >> 5;
    const int wavesPerBlock = blockDim.x >> 5;
    const int gw            = blockIdx.x * wavesPerBlock + waveInBlock;
    const int nWaves        = gridDim.x * wavesPerBlock;

    float ce_acc = 0.0f;   // -t*log(o) contributions (nonzero on at most one lane/row)
    float sl_acc = 0.0f;   // smooth-L1 contributions (lanes 0-3)

    for (int r = gw; r < nRows; r += nWaves) {   // trip count uniform per wave
        float t = 0.0f, o = 0.0f;
        if (lane < C_CLS) {
            t = tgt_s[(size_t)r * C_CLS + lane];   // coalesced 84B burst
            o = out_s[(size_t)r * C_CLS + lane];
        }
        // Butterfly: row-sum of o in all lanes + max-reduce to find fg label lane.
        float osum = o;
        int   fl   = (lane >= 1 && t == 1.0f) ? lane : -1;  // fg uses classes 1..20
        #pragma unroll
        for (int off = 16; off >= 1; off >>= 1) {
            osum += __shfl_xor(osum, off);
            fl    = max(fl, __shfl_xor(fl, off));
        }
        // Cross-entropy: normalize, clip, log only where t != 0 (t is one-hot;
        // t==0 terms are exactly 0 in the reference too since log stays finite).
        if (t != 0.0f) {
            float oc = o / osum;
            oc = fminf(fmaxf(oc, EPS_F), 1.0f - EPS_F);
            ce_acc -= t * logf(oc);
        }
        // Regression: masked elements are the 4 deltas of the fg class.
        if (fl >= 0) {                 // uniform across the wave after reduction
            float v = 0.0f;
            if (lane < 8) {            // lanes 0-3: output deltas, 4-7: target deltas
                const float* p = (lane < 4) ? out_d : tgt_d;
                v = p[(size_t)r * D_DIM + 4 * fl + (lane & 3)];  // 16B-aligned bursts
            }
            float w = __shfl_xor(v, 4);
            if (lane < 4) {
                float dd = fabsf(v - w);
                sl_acc += (dd < 1.0f) ? 0.5f * dd * dd : dd - 0.5f;
            }
        }
    }

    // Fully converged here: WMMA-based exact wave reductions.
    float wave_ce = wave_sum_wmma(ce_acc);
    float wave_sl = wave_sum_wmma(sl_acc);

    __shared__ float s_ce[32];
    __shared__ float s_sl[32];
    if (lane == 0) { s_ce[waveInBlock] = wave_ce; s_sl[waveInBlock] = wave_sl; }
    __syncthreads();
    if (threadIdx.x == 0) {
        float bce = 0.0f, bsl = 0.0f;
        for (int w = 0; w < wavesPerBlock; ++w) { bce += s_ce[w]; bsl += s_sl[w]; }
        partial[2 * blockIdx.x + 0] = bce;
        partial[2 * blockIdx.x + 1] = bsl;
    }
}

__global__ void rcnn_loss_finalize(const float* __restrict__ partial,
                                   int nPartials,
                                   float* __restrict__ out,
                                   int nRows) {
    __shared__ double s_ce[256];
    __shared__ double s_sl[256];
    double ce = 0.0, sl = 0.0;
    for (int i = threadIdx.x; i < nPartials; i += 256) {
        ce += (double)partial[2 * i + 0];
        sl += (double)partial[2 * i + 1];
    }
    s_ce[threadIdx.x] = ce;
    s_sl[threadIdx.x] = sl;
    __syncthreads();
    #pragma unroll
    for (int s = 128; s > 0; s >>= 1) {
        if ((int)threadIdx.x < s) {
            s_ce[threadIdx.x] += s_ce[threadIdx.x + s];
            s_sl[threadIdx.x] += s_sl[threadIdx.x + s];
        }
        __syncthreads();
    }
    if (threadIdx.x == 0) {
        // Reference quirk: floor(mask/4.0) is elementwise floor of {0,0.25} -> 0,
        // so denom = max(0, EPS) = EPS exactly.
        double denom = (double)fmaxf(0.0f, EPS_F);
        double cls   = s_ce[0] / (double)nRows;
        double reg   = s_sl[0] / denom;
        out[0] = (float)(cls + reg);
    }
}

extern "C" void kernel_launch(void* const* d_in, const int* in_sizes, int n_in,
                              void* d_out, int out_size, void* d_ws, size_t ws_size,
                              hipStream_t stream) {
    const float* tgt_d = (const float*)d_in[0];  // target_deltas  (B,R,84)
    const float* tgt_s = (const float*)d_in[1];  // target_scores  (B,R,21)
    const float* out_d = (const float*)d_in[2];  // output_deltas  (B,R,84)
    const float* out_s = (const float*)d_in[3];  // output_scores  (B,R,21)
    float* out     = (float*)d_out;
    float* partial = (float*)d_ws;               // 2 floats per block

    const int nRows = in_sizes[1] / C_CLS;       // 262144 for the reference shapes

    const int BLOCKS  = 512;                     // 4096 waves -> 64 rows/wave
    const int THREADS = 256;                     // 8 wave32 per block
    rcnn_loss_partial<<<BLOCKS, THREADS, 0, stream>>>(tgt_d, tgt_s, out_d, out_s,
                                                      partial, nRows);
    rcnn_loss_finalize<<<1, 256, 0, stream>>>(partial, BLOCKS, out, nRows);
}